// RNNBackbone_61254823575553
// MI455X (gfx1250) — compile-verified
//
#include <hip/hip_runtime.h>

// Problem constants (from reference): B,T,I,H,O and leaky-integrator decay.
#define B_    128
#define T_    512
#define I_    1024
#define H_    2048
#define O_    512
#define BETA  0.9f
#define TCHUNKS 8
#define TCHUNK  (T_ / TCHUNKS)   // 64 timesteps per chunk

typedef __attribute__((ext_vector_type(2))) float v2f;
typedef __attribute__((ext_vector_type(8))) float v8f;

// ---------------------------------------------------------------------------
// Kernel 1: weighted time-collapse of x.
//   xr[b,i] = sum_t c_t * x[b,t,i],  c_t = (1 - beta^(T-t)) / (1 - beta)
// Bandwidth-bound: streams all 256 MB of x exactly once, float4-coalesced.
// Deterministic: partials per T-chunk go to workspace (no float atomics).
// ---------------------------------------------------------------------------
__global__ void time_collapse(const float* __restrict__ x,
                              float* __restrict__ part) {
    const int b    = blockIdx.x;        // batch
    const int c    = blockIdx.y;        // time chunk
    const int tid  = threadIdx.x;       // 0..255 -> one float4 of I
    const int tlo  = c * TCHUNK;
    const int thi  = tlo + TCHUNK;

    // q = beta^(T - t) at t = thi-1  (exponent = T - thi + 1), built by
    // repeated multiply (uniform across the wave -> SALU-friendly).
    float q = 1.0f;
    for (int e = 0; e < T_ - thi + 1; ++e) q *= BETA;
    const float inv1mb = 1.0f / (1.0f - BETA);

    float4 acc = make_float4(0.f, 0.f, 0.f, 0.f);
    const float4* xb = reinterpret_cast<const float4*>(x)
                     + (size_t)b * T_ * (I_ / 4) + tid;
    #pragma unroll 4
    for (int t = thi - 1; t >= tlo; --t) {
        const float cw = (1.0f - q) * inv1mb;
        float4 v = xb[(size_t)t * (I_ / 4)];
        acc.x += cw * v.x;  acc.y += cw * v.y;
        acc.z += cw * v.z;  acc.w += cw * v.w;
        q *= BETA;
    }
    reinterpret_cast<float4*>(part)[((size_t)c * B_ + b) * (I_ / 4) + tid] = acc;
}

// ---------------------------------------------------------------------------
// Kernel 2: reduce the 8 chunk partials -> xr[B, I]
// ---------------------------------------------------------------------------
__global__ void reduce_parts(const float* __restrict__ part,
                             float* __restrict__ xr) {
    const int idx = blockIdx.x * blockDim.x + threadIdx.x;   // float4 index
    const float4* p = reinterpret_cast<const float4*>(part);
    float4 acc = p[idx];
    #pragma unroll
    for (int c = 1; c < TCHUNKS; ++c) {
        float4 v = p[(size_t)c * (B_ * I_ / 4) + idx];
        acc.x += v.x; acc.y += v.y; acc.z += v.z; acc.w += v.w;
    }
    reinterpret_cast<float4*>(xr)[idx] = acc;
}

// ---------------------------------------------------------------------------
// Kernel 3: D[M,N] = A[M,K] @ W[K,N] + bias_scale * bias[N]
// One wave32 per 16x16 output tile using V_WMMA_F32_16X16X4_F32.
// Lane layouts per CDNA5 ISA 7.12.2:
//   A 16x4 : lane L holds A[L&15][k + 2*(L>>4) + j] in vgpr j  (j=0,1)
//   B 4x16 : lane L holds B[k + 2*(L>>4) + j][n0 + (L&15)] in vgpr j
//   C/D    : vgpr r -> row (r + 8*(L>>4)), col n0 + (L&15)
// Block = 256 threads = 8 waves; wave w owns rows [16w, 16w+16) -> full M=128.
// ---------------------------------------------------------------------------
template <int M, int N, int K>
__global__ void wmma_gemm_bias(const float* __restrict__ A,
                               const float* __restrict__ W,
                               const float* __restrict__ bias,
                               float bias_scale,
                               float* __restrict__ D) {
    const int lane  = threadIdx.x & 31;
    const int wave  = threadIdx.x >> 5;          // 0..7 -> m tile
    const int n0    = blockIdx.x * 16;
    const int m0    = wave * 16;
    const int mrow  = lane & 15;                 // A row / B,n column lane
    const int khalf = (lane >> 4) << 1;          // 0 or 2

    v8f acc = {};
    const float* arow = A + (size_t)(m0 + mrow) * K + khalf;
    const float* wcol = W + (size_t)khalf * N + n0 + mrow;

    #pragma unroll 4
    for (int k = 0; k < K; k += 4) {
        // 8-byte aligned vector load of A[m][k+khalf .. k+khalf+1]
        v2f a = *reinterpret_cast<const v2f*>(arow + k);
        v2f b;
        b.x = wcol[(size_t)k * N];
        b.y = wcol[(size_t)(k + 1) * N];
        acc = __builtin_amdgcn_wmma_f32_16x16x4_f32(
                  /*neg_a=*/false, a, /*neg_b=*/false, b,
                  /*c_mod=*/(short)0, acc,
                  /*reuse_a=*/false, /*reuse_b=*/false);
    }

    const int   ncol  = n0 + mrow;
    const float bv    = bias_scale * bias[ncol];
    const int   mbase = m0 + ((lane >> 4) << 3);
    #pragma unroll
    for (int r = 0; r < 8; ++r) {
        D[(size_t)(mbase + r) * N + ncol] = acc[r] + bv;
    }
}

// ---------------------------------------------------------------------------
// Host-side launcher.
// d_in order: x, W_in, b_in, W_out, b_out (all float32). d_out: [B,O] float32.
// Workspace: part (8*B*I) | xr (B*I) | g (B*H)  => ~5.5 MB of f32.
// ---------------------------------------------------------------------------
extern "C" void kernel_launch(void* const* d_in, const int* in_sizes, int n_in,
                              void* d_out, int out_size, void* d_ws, size_t ws_size,
                              hipStream_t stream) {
    (void)in_sizes; (void)n_in; (void)out_size; (void)ws_size;

    const float* x     = (const float*)d_in[0];
    const float* W_in  = (const float*)d_in[1];
    const float* b_in  = (const float*)d_in[2];
    const float* W_out = (const float*)d_in[3];
    const float* b_out = (const float*)d_in[4];
    float* out = (float*)d_out;

    float* ws   = (float*)d_ws;
    float* part = ws;                                   // TCHUNKS * B * I
    float* xr   = part + (size_t)TCHUNKS * B_ * I_;     // B * I
    float* g    = xr   + (size_t)B_ * I_;               // B * H

    // csum = sum_t c_t = sum_{j=1..T} (1 - beta^j) / (1 - beta), in double.
    double cs = 0.0, q = 1.0;
    for (int j = 1; j <= T_; ++j) { q *= (double)BETA; cs += (1.0 - q); }
    cs /= (1.0 - (double)BETA);
    const float csum = (float)cs;

    // 1) weighted time reduction of x  (the 256 MB / ~11 us HBM-bound part)
    time_collapse<<<dim3(B_, TCHUNKS), 256, 0, stream>>>(x, part);
    // 2) combine chunk partials -> xr[B, I]
    reduce_parts<<<dim3((B_ * I_ / 4) / 256), 256, 0, stream>>>(part, xr);
    // 3) g = xr @ W_in + csum * b_in          (128 x 1024 x 2048)
    wmma_gemm_bias<B_, H_, I_><<<H_ / 16, 256, 0, stream>>>(xr, W_in, b_in, csum, g);
    // 4) out = g @ W_out + T * b_out          (128 x 2048 x 512)
    wmma_gemm_bias<B_, O_, H_><<<O_ / 16, 256, 0, stream>>>(g, W_out, b_out, (float)T_, out);
}